// MDNLayer_25039659336011
// MI455X (gfx1250) — compile-verified
//
#include <hip/hip_runtime.h>

typedef float v2f __attribute__((ext_vector_type(2)));
typedef float v8f __attribute__((ext_vector_type(8)));

#define B_SZ   32
#define D_SZ   1024
#define KMIX   16
#define N_CH   302
#define N_POS  394
#define UNITS  (N_CH * KMIX * 3)   /* 14496 = 906 * 16 */

// ---------------------------------------------------------------------------
// Kernel 1: y[b][u] = sum_k x[b][k] * w[k][u] + bias[u]
// Full-f32 GEMM via V_WMMA_F32_16X16X4_F32; one wave32 per 16x16 output tile.
// A layout (ISA 7.12.2, 32-bit A 16x4): lanes 0-15 -> M=lane, v0=K0 v1=K1;
// lanes 16-31 -> M=lane-16, v0=K2 v1=K3.  B mirrors with N=lane&15.
// C/D layout: VGPR r -> M = r + (lane>=16 ? 8 : 0), N = lane&15.
// ---------------------------------------------------------------------------
__global__ __launch_bounds__(32)
void mdn_gemm_wmma(const float* __restrict__ x,
                   const float* __restrict__ w,
                   const float* __restrict__ bias,
                   float* __restrict__ y)
{
    const int lane  = threadIdx.x;        // 0..31
    const int n0    = blockIdx.x * 16;    // 906 column tiles (exact)
    const int m0    = blockIdx.y * 16;    // 2 row tiles (exact)
    const int nlane = lane & 15;
    const int khalf = (lane >> 4) * 2;    // 0 for lanes 0-15, 2 for 16-31

    const float* xrow = x + (size_t)(m0 + nlane) * D_SZ + khalf;
    const float* wcol = w + n0 + nlane;

    v8f acc = {};
    #pragma unroll 8
    for (int k0 = 0; k0 < D_SZ; k0 += 4) {
        // A: contiguous pair from this lane's x row (8B aligned: k0+khalf even)
        v2f a = *(const v2f*)(xrow + k0);
        // B: two rows of w (stride UNITS), same column n0+nlane
        const float* wp = wcol + (size_t)(k0 + khalf) * UNITS;
        v2f bm;
        bm.x = wp[0];
        bm.y = wp[UNITS];
        acc = __builtin_amdgcn_wmma_f32_16x16x4_f32(
                  /*neg_a=*/false, a, /*neg_b=*/false, bm,
                  /*c_mod=*/(short)0, acc, /*reuse_a=*/false, /*reuse_b=*/false);
    }

    const int   col   = n0 + nlane;
    const float bv    = bias[col];
    const int   mbase = m0 + (lane >> 4) * 8;
    #pragma unroll
    for (int r = 0; r < 8; ++r)
        y[(size_t)(mbase + r) * UNITS + col] = acc[r] + bv;
}

// ---------------------------------------------------------------------------
// Kernel 2: per (batch, channel): softmax over 16 pi logits, then
// out[b,c,p] = sum_k w_k * inv_sqrt_2pi/sig_k * exp(-0.5*((pos_p-mu_k)/sig_k)^2)
// Grid (302, 32), 128 threads; params staged in LDS then held in registers.
// ---------------------------------------------------------------------------
__global__ __launch_bounds__(128)
void mdn_mixture(const float* __restrict__ y, float* __restrict__ out)
{
    const int c   = blockIdx.x;   // 0..301
    const int b   = blockIdx.y;   // 0..31
    const int tid = threadIdx.x;

    __shared__ float s_mu[KMIX];
    __shared__ float s_is[KMIX];  // 1/sigma
    __shared__ float s_cf[KMIX];  // softmax_w * inv_sqrt_2pi / sigma
    __shared__ float s_t[KMIX + 2];

    const float* yb = y + (size_t)b * UNITS + c;
    if (tid < KMIX) {
        float mu = yb[(0 * KMIX + tid) * N_CH];
        float sg = yb[(1 * KMIX + tid) * N_CH];
        float pi = yb[(2 * KMIX + tid) * N_CH];
        s_mu[tid] = mu;
        s_is[tid] = __builtin_amdgcn_rcpf(sg);
        s_t[tid]  = pi;
    }
    __syncthreads();
    if (tid == 0) {
        float m = s_t[0];
        #pragma unroll
        for (int k = 1; k < KMIX; ++k) m = fmaxf(m, s_t[k]);
        s_t[KMIX] = m;
    }
    __syncthreads();
    if (tid < KMIX)
        s_t[tid] = __builtin_amdgcn_exp2f((s_t[tid] - s_t[KMIX]) * 1.4426950408889634f);
    __syncthreads();
    if (tid == 0) {
        float s = 0.0f;
        #pragma unroll
        for (int k = 0; k < KMIX; ++k) s += s_t[k];
        s_t[KMIX + 1] = __builtin_amdgcn_rcpf(s);
    }
    __syncthreads();
    if (tid < KMIX)
        s_cf[tid] = s_t[tid] * s_t[KMIX + 1] * 0.3989422804014327f * s_is[tid];
    __syncthreads();

    // Pull mixture params into registers; fold -0.5*log2(e)*inv_sig^2 into q_k.
    float rmu[KMIX], rq[KMIX], rcf[KMIX];
    #pragma unroll
    for (int k = 0; k < KMIX; ++k) {
        float is = s_is[k];
        rmu[k] = s_mu[k];
        rq[k]  = -0.7213475204444817f * is * is;  // -0.5 * log2(e) * is^2
        rcf[k] = s_cf[k];
    }

    float* op = out + ((size_t)b * N_CH + c) * N_POS;
    const float step = 1.0f / (float)(N_POS - 1);
    for (int p = tid; p < N_POS; p += blockDim.x) {
        float pos = (float)p * step;
        float acc = 0.0f;
        #pragma unroll
        for (int k = 0; k < KMIX; ++k) {
            float t = pos - rmu[k];
            acc = fmaf(rcf[k], __builtin_amdgcn_exp2f(rq[k] * t * t), acc);
        }
        op[p] = acc;
    }
}

// ---------------------------------------------------------------------------
extern "C" void kernel_launch(void* const* d_in, const int* in_sizes, int n_in,
                              void* d_out, int out_size, void* d_ws, size_t ws_size,
                              hipStream_t stream)
{
    (void)in_sizes; (void)n_in; (void)out_size; (void)ws_size;

    const float* x    = (const float*)d_in[0];  // (32, 1024)
    const float* w    = (const float*)d_in[1];  // (1024, 14496)
    const float* bias = (const float*)d_in[2];  // (14496,)
    float*       yws  = (float*)d_ws;           // (32, 14496) scratch, 1.86 MB
    float*       out  = (float*)d_out;          // (32, 302, 394, 1)

    dim3 g1(UNITS / 16, B_SZ / 16);             // 906 x 2 tiles
    mdn_gemm_wmma<<<g1, 32, 0, stream>>>(x, w, bias, yws);

    dim3 g2(N_CH, B_SZ);                        // 302 x 32 workgroups
    mdn_mixture<<<g2, 128, 0, stream>>>(yws, out);
}